// PreprocessLayer_8186207667017
// MI455X (gfx1250) — compile-verified
//
#include <hip/hip_runtime.h>

typedef float f4  __attribute__((ext_vector_type(4)));
typedef float v2f __attribute__((ext_vector_type(2)));
typedef float v8f __attribute__((ext_vector_type(8)));

static constexpr unsigned kNPairs    = 16777216u;
static constexpr unsigned kTotF4     = kNPairs * 3u / 4u;          // 12,582,912 float4s of diff
static constexpr int      kRedBlocks = 2048;
static constexpr int      kRedTpb    = 256;
static constexpr int      kRedIters  = (int)(kTotF4 / (unsigned)(kRedBlocks * kRedTpb)); // 24, exact
static constexpr int      kMainTpb   = 256;
static constexpr int      kMainBlocks = (int)(kNPairs / 4u) / kMainTpb;                  // 16384, exact

// Wave32 all-reduce (sum) using V_WMMA_F32_16X16X4_F32.
// A-layout (16x4 f32): lanes 0-15 -> {K0,K1}, lanes 16-31 -> {K2,K3}, M = lane%16.
// With a = {p, 0} and B = ones: D[m][n] = p_m + p_{m+16} for every n.
// C/D layout: lane<16 VGPR v = D[v][lane]; lane>=16 VGPR v = D[v+8][lane-16].
// So sum of the 8 D registers gives half-sum; shfl_xor(16) completes it.
// EXEC must be all ones at this call site (callers have no divergence).
__device__ __forceinline__ float wave_allreduce_wmma(float p) {
  v2f a; a[0] = p;    a[1] = 0.0f;
  v2f b; b[0] = 1.0f; b[1] = 1.0f;
  v8f c = {};
  v8f d = __builtin_amdgcn_wmma_f32_16x16x4_f32(
      /*neg_a=*/false, a, /*neg_b=*/false, b,
      /*c_mod=*/(short)0, c, /*reuse_a=*/false, /*reuse_b=*/false);
  float t = ((d[0] + d[1]) + (d[2] + d[3])) + ((d[4] + d[5]) + (d[6] + d[7]));
  t += __shfl_xor(t, 16, 32);
  return t;
}

// Pass 1: per-block partial sums of squares of diff (read as float4s, fully coalesced,
// exact trip count -> no divergence, EXEC all-1s for the WMMA reduce).
__global__ __launch_bounds__(kRedTpb) void sumsq_partials_kernel(
    const f4* __restrict__ diff4, float* __restrict__ partials) {
  unsigned tid = blockIdx.x * (unsigned)kRedTpb + threadIdx.x;
  float acc = 0.0f;
#pragma unroll
  for (int it = 0; it < kRedIters; ++it) {
    f4 v = diff4[tid + (unsigned)it * (unsigned)(kRedBlocks * kRedTpb)];
    acc = fmaf(v[0], v[0], acc);
    acc = fmaf(v[1], v[1], acc);
    acc = fmaf(v[2], v[2], acc);
    acc = fmaf(v[3], v[3], acc);
  }
  float wsum = wave_allreduce_wmma(acc);

  __shared__ float smem[kRedTpb / 32];
  unsigned lane = threadIdx.x & 31u;
  unsigned wv   = threadIdx.x >> 5;
  if (lane == 0) smem[wv] = wsum;
  __syncthreads();
  if (threadIdx.x == 0) {
    float s = 0.0f;
#pragma unroll
    for (int w = 0; w < kRedTpb / 32; ++w) s += smem[w];   // fixed order -> deterministic
    partials[blockIdx.x] = s;
  }
}

// Pass 2: deterministic final reduce of 2048 partials; writes rnorm = 1/sqrt(sum).
__global__ __launch_bounds__(256) void finalize_kernel(
    const float* __restrict__ partials, float* __restrict__ rnorm_out) {
  __shared__ double smem[256];
  double s = 0.0;
  for (int i = (int)threadIdx.x; i < kRedBlocks; i += 256) s += (double)partials[i];
  smem[threadIdx.x] = s;
  __syncthreads();
  if (threadIdx.x == 0) {
    double t = 0.0;
    for (int i = 0; i < 256; ++i) t += smem[i];            // fixed order -> deterministic
    rnorm_out[0] = (float)(1.0 / sqrt(t));
  }
}

// Pass 3: 4 pairs per thread. 3x b128 loads, 8x b128 non-temporal stores.
__global__ __launch_bounds__(kMainTpb) void main_kernel(
    const f4* __restrict__ diff4, const float* __restrict__ rnorm_p,
    f4* __restrict__ out_nd, f4* __restrict__ out_p5) {
  const float r = rnorm_p[0];
  size_t i = (size_t)blockIdx.x * kMainTpb + threadIdx.x;   // [0, 4194304)

  const f4* in = diff4 + i * 3;
  f4 d0 = in[0];
  f4 d1 = in[1];
  f4 d2 = in[2];
  f4 n0 = d0 * r;
  f4 n1 = d1 * r;
  f4 n2 = d2 * r;

  f4* nd = out_nd + i * 3;
  __builtin_nontemporal_store(n0, nd + 0);
  __builtin_nontemporal_store(n1, nd + 1);
  __builtin_nontemporal_store(n2, nd + 2);

  const float T  = 1.0f / 3.0f;
  const float T2 = 2.0f / 3.0f;
  // pair j components (x,y,z) from the 12 normalized floats
  float px[4] = { n0[0], n0[3], n1[2], n2[1] };
  float py[4] = { n0[1], n1[0], n1[3], n2[2] };
  float pz[4] = { n0[2], n1[1], n2[0], n2[3] };
  float o[20];
#pragma unroll
  for (int j = 0; j < 4; ++j) {
    float x = px[j], y = py[j], z = pz[j];
    float x2 = x * x, y2 = y * y, z2 = z * z;
    o[5 * j + 0] = T2 * x2 - T * y2 - T * z2;
    o[5 * j + 1] = T2 * y2 - T * x2 - T * z2;
    o[5 * j + 2] = x * y;
    o[5 * j + 3] = x * z;
    o[5 * j + 4] = y * z;
  }

  f4* p5 = out_p5 + i * 5;
#pragma unroll
  for (int q = 0; q < 5; ++q) {
    f4 v;
    v[0] = o[4 * q + 0];
    v[1] = o[4 * q + 1];
    v[2] = o[4 * q + 2];
    v[3] = o[4 * q + 3];
    __builtin_nontemporal_store(v, p5 + q);
  }
}

extern "C" void kernel_launch(void* const* d_in, const int* in_sizes, int n_in,
                              void* d_out, int out_size, void* d_ws, size_t ws_size,
                              hipStream_t stream) {
  (void)in_sizes; (void)n_in; (void)out_size; (void)ws_size;

  const f4* diff4   = (const f4*)d_in[0];   // diff: (16777216, 3) float32, flat
  float*    partials = (float*)d_ws;        // 2048 floats
  float*    rnorm    = partials + kRedBlocks; // 1 float

  f4* out_nd = (f4*)d_out;                  // norm_diff: 3*N floats = kTotF4 float4s
  f4* out_p5 = out_nd + (size_t)kTotF4;     // diff_p5:  5*N floats, starts 16B-aligned

  sumsq_partials_kernel<<<kRedBlocks, kRedTpb, 0, stream>>>(diff4, partials);
  finalize_kernel<<<1, 256, 0, stream>>>(partials, rnorm);
  main_kernel<<<kMainBlocks, kMainTpb, 0, stream>>>(diff4, rnorm, out_nd, out_p5);
}